// PrimitiveGate_24842090840148
// MI455X (gfx1250) — compile-verified
//
#include <hip/hip_runtime.h>

// D = A(16x4) * B(4x16) + C, f32 WMMA on gfx1250 (wave32).
typedef __attribute__((ext_vector_type(2))) float v2f;
typedef __attribute__((ext_vector_type(8))) float v8f;

#define S1 (1u << 20)   // qubit-5 stride (elements): bit 20 of the flat index
#define S2 (1u << 12)   // qubit-13 stride (elements): bit 12 of the flat index

// Expand a 24-bit column id into a 26-bit element index by inserting zero
// bits at positions 12 (qubit 13) and 20 (qubit 5).
__device__ __forceinline__ unsigned expand_idx(unsigned c) {
    unsigned b = (c & 0xFFFu)   | ((c >> 12) << 13);  // insert 0 at bit 12
    return     (b & 0xFFFFFu)   | ((b >> 20) << 21);  // insert 0 at bit 20
}

__global__ __launch_bounds__(256)
void gate_wmma_kernel(const float* __restrict__ state,
                      const float* __restrict__ mat,   // 4x4 row-major gate
                      float* __restrict__ out)
{
    const int tid  = threadIdx.x;
    const int lane = tid & 31;
    const int wv   = tid >> 5;       // wave id within block (8 waves)
    const int half = lane >> 4;      // 0: lanes 0-15 (K=0,1) | 1: lanes 16-31 (K=2,3)
    const int lcol = lane & 15;

    // Each wave owns 32 columns: c0 .. c0+31.
    const unsigned c0 = (blockIdx.x * 8u + (unsigned)wv) * 32u;

    // ---- Gate into A operands (A layout: lane = M row, VGPR = K within half) ----
    const int kb = half * 2;         // K base for this lane half
    // A1: gate occupies rows M = 0..3, everything else zero.
    int r1  = lcol;
    int r1c = (r1 < 3) ? r1 : 3;                 // clamp -> always in-bounds load
    float g10 = mat[r1c * 4 + kb];
    float g11 = mat[r1c * 4 + kb + 1];
    v2f a1;
    a1.x = (r1 < 4) ? g10 : 0.0f;
    a1.y = (r1 < 4) ? g11 : 0.0f;
    // A2: gate occupies rows M = 8..11, everything else zero.
    int r2  = lcol - 8;
    int r2c = (r2 < 0) ? 0 : ((r2 > 3) ? 3 : r2);
    float g20 = mat[r2c * 4 + kb];
    float g21 = mat[r2c * 4 + kb + 1];
    bool in2 = (r2 >= 0) && (r2 < 4);
    v2f a2;
    a2.x = in2 ? g20 : 0.0f;
    a2.y = in2 ? g21 : 0.0f;

    // ---- State columns into B operands ----
    // B layout: lane L<16 holds K=0 (v0) / K=1 (v1) of column (c + L);
    //           lane L>=16 holds K=2 / K=3 of column (c + L-16).
    // K offsets: K=0 -> 0, K=1 -> S2, K=2 -> S1, K=3 -> S1+S2.
    unsigned offB1 = expand_idx(c0 + (unsigned)lcol)       + (unsigned)half * S1;
    unsigned offB2 = expand_idx(c0 + 16u + (unsigned)lcol) + (unsigned)half * S1;
    v2f b1, b2;
    b1.x = state[offB1];
    b1.y = state[offB1 + S2];
    b2.x = state[offB2];
    b2.y = state[offB2 + S2];

    // ---- Two chained WMMAs merge both 16-column tiles into one D ----
    // A1 rows 8-15 are zero and A2 rows 0-7 are zero, so accumulating the
    // second WMMA on top of the first leaves: D VGPR v (v=0..3) =
    //   lanes 0-15  -> out[row v, col c0+lane]      (M = v,   from WMMA#1)
    //   lanes 16-31 -> out[row v, col c0+lane]      (M = v+8, from WMMA#2)
    v8f acc = {};
    acc = __builtin_amdgcn_wmma_f32_16x16x4_f32(false, a1, false, b1,
                                                (short)0, acc, false, false);
    acc = __builtin_amdgcn_wmma_f32_16x16x4_f32(false, a2, false, b2,
                                                (short)0, acc, false, false);

    // ---- Store: 4 row streams, each a single 128B-coalesced store ----
    unsigned ob = expand_idx(c0 + (unsigned)lane);
    out[ob]           = acc[0];   // row 0 -> offset 0
    out[ob + S2]      = acc[1];   // row 1 -> +S2
    out[ob + S1]      = acc[2];   // row 2 -> +S1
    out[ob + S1 + S2] = acc[3];   // row 3 -> +S1+S2
}

extern "C" void kernel_launch(void* const* d_in, const int* in_sizes, int n_in,
                              void* d_out, int out_size, void* d_ws, size_t ws_size,
                              hipStream_t stream) {
    (void)in_sizes; (void)n_in; (void)out_size; (void)d_ws; (void)ws_size;
    const float* state = (const float*)d_in[0];   // 2^26 f32
    const float* mat   = (const float*)d_in[1];   // 16 f32 (4x4 row-major)
    float*       out   = (float*)d_out;           // 2^26 f32

    // 2^24 columns total; 256 threads = 8 waves = 256 columns per block.
    dim3 grid(65536), block(256);
    hipLaunchKernelGGL(gate_wmma_kernel, grid, block, 0, stream, state, mat, out);
}